// FourierShellCorrelation_21320217658067
// MI455X (gfx1250) — compile-verified
//
#include <hip/hip_runtime.h>
#include <hip/hip_bf16.h>
#include <math.h>

// ---------------------------------------------------------------------------
// FSC(X, Y) on MI455X / gfx1250.
// 3D FFT = 3 passes of (DFT-as-GEMM, M=65536 x N=256 x K=256) using
// V_WMMA_F32_16X16X4_F32. Twiddles pre-generated in WMMA A-fragment layout.
// Each pass writes out[k][r] so the next axis becomes contiguous.
// LDS staging uses GLOBAL_LOAD_ASYNC_TO_LDS_B128 (ASYNCcnt) when available.
// ---------------------------------------------------------------------------

typedef float v2f __attribute__((ext_vector_type(2)));
typedef float v8f __attribute__((ext_vector_type(8)));

#define VOXELS   16777216u      // 256^3
#define NROWS    65536          // 256^2 rows per pass
#define NBLK     32             // rows (r) per workgroup
#define LSTRIDE  132            // LDS row stride (floats): bank = (4r+l)%64, conflict-free
#define KSTAGE   128            // K staged per phase (2 phases of 32 k-steps)
#define NSHELL   222            // floor(sqrt(3*128^2)) + 1
#define SHPAD    224

#if defined(__has_builtin)
#if __has_builtin(__builtin_amdgcn_global_load_async_to_lds_b128) && \
    __has_builtin(__builtin_amdgcn_s_wait_asynccnt)
#define USE_ASYNC_LDS 1
#endif
#endif
#ifndef USE_ASYNC_LDS
#define USE_ASYNC_LDS 0
#endif

#if USE_ASYNC_LDS
// Parameter types per hipcc diagnostic: v4i in AS(1) (printed "__device__")
// for the global source, AS(3) for the LDS destination.
typedef int v4i_vs __attribute__((vector_size(16)));
typedef __attribute__((address_space(1))) v4i_vs glb_v4i;
typedef __attribute__((address_space(3))) v4i_vs lds_v4i;
#endif

static __device__ inline v8f wmma4(v2f a, v2f b, v8f c) {
  // D = A(16x4, f32) * B(4x16, f32) + C(16x16, f32)
  return __builtin_amdgcn_wmma_f32_16x16x4_f32(
      /*neg_a=*/false, a, /*neg_b=*/false, b,
      /*c_mod=*/(short)0, c, /*reuse_a=*/false, /*reuse_b=*/false);
}

// ---------------------------------------------------------------------------
// Twiddle fragments. W[m,l] = e^{-2*pi*i*m*l/256} = c + i*mneg, mneg = -sin.
// Stored per variant v (0=cos, 1=+sin, 2=-sin) in A-fragment layout:
//   tw[v][mt(16)][kstep(64)][lane(32)][j(2)]
// lane L: m = mt*16 + (L&15), l = kstep*4 + 2*(L>>4) + j   (ISA 16x4 A layout)
// ---------------------------------------------------------------------------
__global__ __launch_bounds__(256) void gen_twiddles(float* __restrict__ tw) {
  int idx = blockIdx.x * 256 + threadIdx.x;
  if (idx >= 3 * 16 * 64 * 64) return;
  int j     =  idx        & 1;
  int lane  = (idx >> 1)  & 31;
  int kstep = (idx >> 6)  & 63;
  int mt    = (idx >> 12) & 15;
  int v     =  idx >> 16;
  int m = mt * 16 + (lane & 15);
  int l = kstep * 4 + ((lane >> 4) << 1) + j;
  float th = 6.28318530717958647692f * (float)((m * l) & 255) * (1.0f / 256.0f);
  float s = sinf(th), c = cosf(th);
  tw[idx] = (v == 0) ? c : ((v == 1) ? s : -s);
}

__global__ __launch_bounds__(256) void zero_sums(float* __restrict__ sums) {
  for (int i = threadIdx.x; i < 2 * 3 * SHPAD; i += 256) sums[i] = 0.0f;
}

// ---------------------------------------------------------------------------
// One DFT pass: dst[k*65536 + r] = sum_l W[k,l] * src[r*256 + l]   (complex)
// Block: 256 threads (8 waves), covers 32 rows x all 256 k.
// Wave w: k in [32w, 32w+32) (2 M-tiles) x 32 rows (2 N-tiles).
// A = twiddle fragments (global, L2-resident), B = input from padded LDS.
// ---------------------------------------------------------------------------
template <bool CPLX>
__global__ __launch_bounds__(256) void dft_pass(
    const float* __restrict__ srcRe, const float* __restrict__ srcIm,
    float* __restrict__ dstRe, float* __restrict__ dstIm,
    const float* __restrict__ tw) {
  extern __shared__ float lds[];               // [CPLX?2:1][NBLK][LSTRIDE]
  const int rbase = blockIdx.x * NBLK;
  const int t = threadIdx.x;
  const int lane = t & 31;
  const int wv = t >> 5;

  float* ldsRe = lds;
  float* ldsIm = lds + NBLK * LSTRIDE;         // only touched when CPLX

  const float* twc = tw;                        // cos
  const float* twp = tw + 16 * 64 * 64;         // +sin
  const float* twm = tw + 2 * 16 * 64 * 64;     // -sin

  v8f accRe[2][2], accIm[2][2];
  const v8f vz = {0.f, 0.f, 0.f, 0.f, 0.f, 0.f, 0.f, 0.f};
#pragma unroll
  for (int mi = 0; mi < 2; ++mi)
#pragma unroll
    for (int nj = 0; nj < 2; ++nj) { accRe[mi][nj] = vz; accIm[mi][nj] = vz; }

  for (int phase = 0; phase < 2; ++phase) {
    __syncthreads();  // previous-stage readers done before overwrite
    // Stage 32 rows x 128 floats per component into LDS (16B per lane).
#pragma unroll
    for (int i = 0; i < 4; ++i) {
      int g  = i * 256 + t;      // 0..1023
      int r  = g >> 5;           // 32 float4 per row
      int l  = (g & 31) << 2;
      size_t goff = (size_t)(rbase + r) * 256 + (size_t)phase * KSTAGE + l;
#if USE_ASYNC_LDS
      // Async DMA: global -> LDS, no VGPR round trip; tracked by ASYNCcnt.
      __builtin_amdgcn_global_load_async_to_lds_b128(
          (glb_v4i*)(srcRe + goff), (lds_v4i*)(ldsRe + r * LSTRIDE + l), 0, 0);
      if constexpr (CPLX)
        __builtin_amdgcn_global_load_async_to_lds_b128(
            (glb_v4i*)(srcIm + goff), (lds_v4i*)(ldsIm + r * LSTRIDE + l), 0, 0);
#else
      float4 a = *(const float4*)(srcRe + goff);
      *(float4*)(ldsRe + r * LSTRIDE + l) = a;
      if constexpr (CPLX) {
        float4 b = *(const float4*)(srcIm + goff);
        *(float4*)(ldsIm + r * LSTRIDE + l) = b;
      }
#endif
      if (phase == 0) {
        __builtin_prefetch(srcRe + goff + KSTAGE, 0, 1);
        if constexpr (CPLX) __builtin_prefetch(srcIm + goff + KSTAGE, 0, 1);
      }
    }
#if USE_ASYNC_LDS
    __builtin_amdgcn_s_wait_asynccnt(0);   // this wave's DMA into LDS complete
#endif
    __syncthreads();

    for (int ks = 0; ks < 32; ++ks) {
      const int kstep = phase * 32 + ks;
      const int lloc  = ks * 4 + ((lane >> 4) << 1);   // B-fragment l (j=0)
      // B fragments: B[l, n=r] = src[r, l] from LDS (conflict-free, 8B loads)
      v2f br[2], bi[2];
#pragma unroll
      for (int nj = 0; nj < 2; ++nj) {
        int rl = nj * 16 + (lane & 15);
        br[nj] = *(const v2f*)(ldsRe + rl * LSTRIDE + lloc);
        if constexpr (CPLX)
          bi[nj] = *(const v2f*)(ldsIm + rl * LSTRIDE + lloc);
      }
#pragma unroll
      for (int mi = 0; mi < 2; ++mi) {
        const int mt = wv * 2 + mi;
        const size_t off = ((size_t)mt * 64 + kstep) * 64 + lane * 2;
        v2f ac = *(const v2f*)(twc + off);
        v2f am = *(const v2f*)(twm + off);
        // out_re = c*ar + (+sin)*ai ; out_im = c*ai + (-sin)*ar
#pragma unroll
        for (int nj = 0; nj < 2; ++nj) {
          accRe[mi][nj] = wmma4(ac, br[nj], accRe[mi][nj]);
          accIm[mi][nj] = wmma4(am, br[nj], accIm[mi][nj]);
        }
        if constexpr (CPLX) {
          v2f ap = *(const v2f*)(twp + off);
#pragma unroll
          for (int nj = 0; nj < 2; ++nj) {
            accRe[mi][nj] = wmma4(ap, bi[nj], accRe[mi][nj]);
            accIm[mi][nj] = wmma4(ac, bi[nj], accIm[mi][nj]);
          }
        }
      }
    }
  }

  // Store: D lane L, vgpr i -> row m = i + 8*(L>>4), col n = L&15.
  // dst addr = k*65536 + r (contiguous in r => 64B half-wave segments).
#pragma unroll
  for (int mi = 0; mi < 2; ++mi) {
    const int kb = wv * 32 + mi * 16 + ((lane >> 4) << 3);
#pragma unroll
    for (int nj = 0; nj < 2; ++nj) {
      const int rg = rbase + nj * 16 + (lane & 15);
#pragma unroll
      for (int i = 0; i < 8; ++i) {
        size_t addr = (size_t)(kb + i) * NROWS + rg;
        dstRe[addr] = accRe[mi][nj][i];
        dstIm[addr] = accIm[mi][nj][i];
      }
    }
  }
}

// ---------------------------------------------------------------------------
// Shell reduction: layout after 3 passes is [kd][kh][kw].
// ---------------------------------------------------------------------------
__global__ __launch_bounds__(256) void shell_reduce(
    const float* __restrict__ fxr, const float* __restrict__ fxi,
    const float* __restrict__ fyr, const float* __restrict__ fyi,
    float* __restrict__ sums /* [3][SHPAD] for this batch */) {
  __shared__ float sn[SHPAD], sp[SHPAD], sq[SHPAD];
  const int t = threadIdx.x;
  for (int s = t; s < SHPAD; s += 256) { sn[s] = 0.f; sp[s] = 0.f; sq[s] = 0.f; }
  __syncthreads();

  const unsigned base = blockIdx.x * 8192u;  // 256 threads * 32 voxels
#pragma unroll 4
  for (int i = 0; i < 32; ++i) {
    unsigned n = base + (unsigned)i * 256u + t;
    int kd = n >> 16, kh = (n >> 8) & 255, kw = n & 255;
    int fd = (kd < 128) ? kd : kd - 256;
    int fh = (kh < 128) ? kh : kh - 256;
    int fw = (kw < 128) ? kw : kw - 256;
    int r2 = fd * fd + fh * fh + fw * fw;
    int ib = (int)sqrtf((float)r2);            // exact integer floor-sqrt
    ib -= (ib * ib > r2);
    ib += ((ib + 1) * (ib + 1) <= r2);
    float xr = fxr[n], xi = fxi[n], yr = fyr[n], yi = fyi[n];
    atomicAdd(&sn[ib], xr * yr + xi * yi);     // ds_add_f32
    atomicAdd(&sp[ib], xr * xr + xi * xi);
    atomicAdd(&sq[ib], yr * yr + yi * yi);
  }
  __syncthreads();
  for (int s = t; s < SHPAD; s += 256) {
    atomicAdd(&sums[s], sn[s]);                // global_atomic_add_f32
    atomicAdd(&sums[SHPAD + s], sp[s]);
    atomicAdd(&sums[2 * SHPAD + s], sq[s]);
  }
}

__global__ __launch_bounds__(256) void fsc_final(const float* __restrict__ sums,
                                                 float* __restrict__ out) {
  __shared__ float red[256];
  const int t = threadIdx.x;
  float acc = 0.f;
  for (int i = t; i < 2 * NSHELL; i += 256) {
    int b = i / NSHELL, s = i - b * NSHELL;
    const float* S = sums + b * 3 * SHPAD;
    acc += S[s] / sqrtf(S[SHPAD + s] * S[2 * SHPAD + s] + 1e-8f);
  }
  red[t] = acc;
  __syncthreads();
  for (int off = 128; off > 0; off >>= 1) {
    if (t < off) red[t] += red[t + off];
    __syncthreads();
  }
  if (t == 0) out[0] = red[0] / (float)(2 * NSHELL);
}

// ---------------------------------------------------------------------------
extern "C" void kernel_launch(void* const* d_in, const int* in_sizes, int n_in,
                              void* d_out, int out_size, void* d_ws, size_t ws_size,
                              hipStream_t stream) {
  (void)in_sizes; (void)n_in; (void)out_size; (void)ws_size;
  const float* X = (const float*)d_in[0];  // [2,1,256,256,256] f32
  const float* Y = (const float*)d_in[1];

  float* ws = (float*)d_ws;
  float* tw   = ws;                 // 3*16*64*64 = 196608 floats (768 KB)
  float* sums = ws + 196608;        // 2*3*224 floats
  float* b0r  = ws + 262144;        // 6 x 64MB volume buffers
  float* b0i  = b0r + VOXELS;
  float* b1r  = b0i + VOXELS;
  float* b1i  = b1r + VOXELS;
  float* fxr  = b1i + VOXELS;
  float* fxi  = fxr + VOXELS;

  const size_t LDS_R = (size_t)NBLK * LSTRIDE * sizeof(float);       // ~16.9 KB
  const size_t LDS_C = 2 * LDS_R;                                    // ~33.8 KB
  const dim3 gGemm(NROWS / NBLK), blk(256);

  gen_twiddles<<<768, 256, 0, stream>>>(tw);
  zero_sums<<<1, 256, 0, stream>>>(sums);

  for (int b = 0; b < 2; ++b) {
    const float* Xb = X + (size_t)b * VOXELS;
    const float* Yb = Y + (size_t)b * VOXELS;
    // Fx = FFT3(Xb): w-axis (real), h-axis, d-axis -> fx
    dft_pass<false><<<gGemm, blk, LDS_R, stream>>>(Xb, nullptr, b0r, b0i, tw);
    dft_pass<true ><<<gGemm, blk, LDS_C, stream>>>(b0r, b0i, b1r, b1i, tw);
    dft_pass<true ><<<gGemm, blk, LDS_C, stream>>>(b1r, b1i, fxr, fxi, tw);
    // Fy = FFT3(Yb) -> b0
    dft_pass<false><<<gGemm, blk, LDS_R, stream>>>(Yb, nullptr, b0r, b0i, tw);
    dft_pass<true ><<<gGemm, blk, LDS_C, stream>>>(b0r, b0i, b1r, b1i, tw);
    dft_pass<true ><<<gGemm, blk, LDS_C, stream>>>(b1r, b1i, b0r, b0i, tw);
    // Shell sums for this batch
    shell_reduce<<<VOXELS / 8192, 256, 0, stream>>>(fxr, fxi, b0r, b0i,
                                                    sums + b * 3 * SHPAD);
  }
  fsc_final<<<1, 256, 0, stream>>>(sums, (float*)d_out);
}